// DFTB_Layer_4922032521596
// MI455X (gfx1250) — compile-verified
//
#include <hip/hip_runtime.h>

// ---------------------------------------------------------------------------
// DFTB pipeline for MI455X (gfx1250, wave32).
// Stages: spline matvec -> rotation gather -> per-molecule assembly (H,F) ->
// segment-sum Erep -> per-molecule Loewdin generalized eigensolve with
// parallel Jacobi + f32 WMMA (v_wmma_f32_16x16x4_f32) GEMMs in LDS.
// ---------------------------------------------------------------------------

#define BS    64
#define LDA   65          // LDS row stride (odd -> bank-conflict-free cols)
#define BLK   256         // 8 waves of 32

typedef __attribute__((ext_vector_type(2))) float v2f;
typedef __attribute__((ext_vector_type(8))) float v8f;

// ---------------------------------------------------------------------------
// Wave-level 64x64(xK) GEMM on LDS buffers using V_WMMA_F32_16X16X4_F32.
// D = alpha * op(A) * op(B), optional column-permutation on B's N index.
// 8 waves split the 4x4 grid of 16x16 output tiles (2 tiles each).
// Fragment layouts per CDNA5 ISA 7.12.2:
//   A 16x4 : lanes 0-15 -> (m, k+0/k+1), lanes 16-31 -> (m, k+2/k+3)
//   B 4x16 : v0 = B[k+2*half][n], v1 = B[k+2*half+1][n], n = lane%16
//   D 16x16: vgpr r -> row (tm + r + 8*half), col tn + lane%16
// ---------------------------------------------------------------------------
template <bool TA, bool TB>
__device__ inline void gemm64(const float* __restrict__ A,
                              const float* __restrict__ B,
                              float* __restrict__ D,
                              int K, float alpha,
                              const int* __restrict__ permB, int tid)
{
    const int lane = tid & 31;
    const int wave = tid >> 5;
    const int half = lane >> 4;    // 0 or 1
    const int l    = lane & 15;

    for (int tile = wave; tile < 16; tile += 8) {
        const int tm = (tile >> 2) << 4;
        const int tn = (tile & 3) << 4;
        const int pn = permB[tn + l];      // permuted N index for B
        v8f acc = {};
#pragma unroll 8
        for (int k = 0; k < K; k += 4) {
            const int ka = k + 2 * half;
            v2f af, bf;
            if (TA) {
                af.x = A[ka * LDA + tm + l];
                af.y = A[(ka + 1) * LDA + tm + l];
            } else {
                af.x = A[(tm + l) * LDA + ka];
                af.y = A[(tm + l) * LDA + ka + 1];
            }
            if (TB) {
                bf.x = B[pn * LDA + ka];
                bf.y = B[pn * LDA + ka + 1];
            } else {
                bf.x = B[ka * LDA + pn];
                bf.y = B[(ka + 1) * LDA + pn];
            }
            acc = __builtin_amdgcn_wmma_f32_16x16x4_f32(
                false, af, false, bf, (short)0, acc, false, false);
        }
#pragma unroll
        for (int r = 0; r < 8; ++r)
            D[(tm + r + 8 * half) * LDA + tn + l] = alpha * acc[r];
    }
}

// ---------------------------------------------------------------------------
// Block-wide parallel-ordering Jacobi eigensolver for 64x64 symmetric matrix
// in LDS.  63 round-robin rounds of 32 disjoint plane rotations per sweep.
// On exit: eigenvalues on diag(Mb), eigenvectors in columns of Vb.
// ---------------------------------------------------------------------------
__device__ inline void jacobi64(float* Mb, float* Vb,
                                int* top, int* bot, int* pp, int* qq,
                                float* cs, float* sn, int tid, int sweeps)
{
    for (int i = tid; i < BS * BS; i += BLK) {
        int r = i >> 6, c = i & 63;
        Vb[r * LDA + c] = (r == c) ? 1.0f : 0.0f;
    }
    if (tid < 32) { top[tid] = 2 * tid; bot[tid] = 2 * tid + 1; }
    __syncthreads();

    for (int sw = 0; sw < sweeps; ++sw) {
        for (int round = 0; round < 63; ++round) {
            // --- compute 32 independent Givens rotations ---
            if (tid < 32) {
                int p = top[tid], q = bot[tid];
                if (p > q) { int t = p; p = q; q = t; }
                pp[tid] = p; qq[tid] = q;
                float apq = Mb[p * LDA + q];
                float c = 1.0f, s = 0.0f;
                if (fabsf(apq) > 1e-20f) {
                    float app = Mb[p * LDA + p];
                    float aqq = Mb[q * LDA + q];
                    float tau = (aqq - app) / (2.0f * apq);
                    float t = ((tau >= 0.0f) ? 1.0f : -1.0f) /
                              (fabsf(tau) + sqrtf(1.0f + tau * tau));
                    c = rsqrtf(1.0f + t * t);
                    s = t * c;
                }
                cs[tid] = c; sn[tid] = s;
            }
            __syncthreads();
            // --- rows: A <- J^T A (pairs are disjoint -> race free) ---
            for (int t = tid; t < 32 * 64; t += BLK) {
                int pair = t >> 6, j = t & 63;
                int p = pp[pair], q = qq[pair];
                float c = cs[pair], s = sn[pair];
                float ap = Mb[p * LDA + j], aq = Mb[q * LDA + j];
                Mb[p * LDA + j] = c * ap - s * aq;
                Mb[q * LDA + j] = s * ap + c * aq;
            }
            __syncthreads();
            // --- cols: A <- A J, V <- V J ---
            for (int t = tid; t < 32 * 128; t += BLK) {
                int pair = t >> 7, idx = t & 127;
                float* buf = (idx < 64) ? Mb : Vb;
                int i = idx & 63;
                int p = pp[pair], q = qq[pair];
                float c = cs[pair], s = sn[pair];
                float aip = buf[i * LDA + p], aiq = buf[i * LDA + q];
                buf[i * LDA + p] = c * aip - s * aiq;
                buf[i * LDA + q] = s * aip + c * aiq;
            }
            __syncthreads();
            // --- rotate tournament (chess round-robin, top[0] fixed) ---
            if (tid == 0) {
                int b0 = bot[0];
                int tl = top[31];
                for (int i = 0; i < 31; ++i) bot[i] = bot[i + 1];
                bot[31] = tl;
                for (int i = 31; i >= 2; --i) top[i] = top[i - 1];
                top[1] = b0;
            }
            __syncthreads();
        }
    }
}

// ---------------------------------------------------------------------------
// Stage 1: net_vals = A @ coeffs + b  (one wave per row, shuffle reduce)
// ---------------------------------------------------------------------------
__global__ __launch_bounds__(BLK)
void spline_kernel(const float* __restrict__ A, const float* __restrict__ b,
                   const float* __restrict__ coeffs,
                   float* __restrict__ net_vals, int nsp)
{
    __shared__ float csh[64];
    int tid = threadIdx.x;
    if (tid < 64) csh[tid] = coeffs[tid];
    __syncthreads();
    int lane = tid & 31, wave = tid >> 5;
    int row = blockIdx.x * 8 + wave;
    if (row >= nsp) return;
    const float* a = A + (size_t)row * 64;
    float s = a[lane] * csh[lane] + a[lane + 32] * csh[lane + 32];
    for (int off = 16; off; off >>= 1) s += __shfl_xor(s, off, 32);
    if (lane == 0) net_vals[row] = s + b[row];
}

// ---------------------------------------------------------------------------
// Stage 2: rot_out = [0,1] ++ net_vals[g_rot_direct] ++ rotated 3-vectors
// ---------------------------------------------------------------------------
__global__ __launch_bounds__(BLK)
void rot_kernel(const float* __restrict__ net_vals,
                const float* __restrict__ rot_tensor,
                const int* __restrict__ g_rot_direct,
                const int* __restrict__ g_rot,
                float* __restrict__ rot_out, int l0, int nrot)
{
    int t = blockIdx.x * blockDim.x + threadIdx.x;
    if (t == 0) { rot_out[0] = 0.0f; rot_out[1] = 1.0f; }
    if (t < l0) {
        rot_out[2 + t] = net_vals[g_rot_direct[t]];
    } else if (t < l0 + nrot) {
        int n = t - l0;
        float v0 = net_vals[g_rot[3 * n + 0]];
        float v1 = net_vals[g_rot[3 * n + 1]];
        float v2 = net_vals[g_rot[3 * n + 2]];
        const float* R = rot_tensor + (size_t)n * 9;
        float* o = rot_out + 2 + l0 + 3 * n;
        o[0] = R[0] * v0 + R[1] * v1 + R[2] * v2;
        o[1] = R[3] * v0 + R[4] * v1 + R[5] * v2;
        o[2] = R[6] * v0 + R[7] * v1 + R[8] * v2;
    }
}

// ---------------------------------------------------------------------------
// Stage 3 (one block per molecule): Mulliken dQ, ep = G@dQ, ener2,
// H = rot_out[g_oper], F = H - 0.5*S*(ep_i + ep_j).
// ---------------------------------------------------------------------------
__global__ __launch_bounds__(BLK)
void mull_kernel(const float* __restrict__ S, const float* __restrict__ G,
                 const float* __restrict__ rho,
                 const float* __restrict__ qneutral,
                 const int* __restrict__ g_oper,
                 const float* __restrict__ rot_out,
                 float* __restrict__ Fws, float* __restrict__ Hws,
                 float* __restrict__ E2)
{
    __shared__ float part[64][4];
    __shared__ float dQ[64], ep[64];
    __shared__ float red2[64];

    const int mol = blockIdx.x, tid = threadIdx.x;
    const size_t mb = (size_t)mol * 4096;
    const float* Sm = S + mb;
    const float* Gm = G + mb;
    const float* Rm = rho + mb;
    const float* qn = qneutral + (size_t)mol * 64;

    const int row = tid >> 2, seg = tid & 3;

    // GOP[i] = sum_j rho*S
    float p = 0.0f;
    for (int j = seg * 16; j < seg * 16 + 16; ++j)
        p += Rm[row * 64 + j] * Sm[row * 64 + j];
    part[row][seg] = p;
    __syncthreads();
    if (tid < 64)
        dQ[tid] = qn[tid] - (part[tid][0] + part[tid][1] + part[tid][2] + part[tid][3]);
    __syncthreads();

    // ep = G @ dQ
    p = 0.0f;
    for (int j = seg * 16; j < seg * 16 + 16; ++j)
        p += Gm[row * 64 + j] * dQ[j];
    part[row][seg] = p;
    __syncthreads();
    if (tid < 64)
        ep[tid] = part[tid][0] + part[tid][1] + part[tid][2] + part[tid][3];
    __syncthreads();

    // ener2 = 0.5 * sum dQ*ep
    if (tid < 64) red2[tid] = dQ[tid] * ep[tid];
    __syncthreads();
    for (int s2 = 32; s2 > 0; s2 >>= 1) {
        if (tid < s2) red2[tid] += red2[tid + s2];
        __syncthreads();
    }
    if (tid == 0) E2[mol] = 0.5f * red2[0];

    // H and F
    const int* go = g_oper + mb;
    for (int i = tid; i < 4096; i += BLK) {
        int r = i >> 6, c = i & 63;
        float h = rot_out[go[i]];
        Hws[mb + i] = h;
        Fws[mb + i] = h - 0.5f * Sm[i] * (ep[r] + ep[c]);
    }
}

// ---------------------------------------------------------------------------
// Stage 4: Erep = segment_sum(net_vals[g_rep], seg_ids).  seg_ids sorted ->
// one wave per segment via binary search.  Deterministic (no atomics).
// ---------------------------------------------------------------------------
__device__ inline int lower_bound_i(const int* a, int n, int key)
{
    int lo = 0, hi = n;
    while (lo < hi) { int mid = (lo + hi) >> 1; if (a[mid] < key) lo = mid + 1; else hi = mid; }
    return lo;
}

__global__ __launch_bounds__(BLK)
void erep_kernel(const float* __restrict__ net_vals,
                 const int* __restrict__ g_rep, const int* __restrict__ seg,
                 float* __restrict__ Erep, int nrep, int nmol)
{
    int gw = (blockIdx.x * blockDim.x + threadIdx.x) >> 5;
    int lane = threadIdx.x & 31;
    if (gw >= nmol) return;
    int lo = lower_bound_i(seg, nrep, gw);
    int hi = lower_bound_i(seg, nrep, gw + 1);
    float s = 0.0f;
    for (int i = lo + lane; i < hi; i += 32) s += net_vals[g_rep[i]];
    for (int off = 16; off; off >>= 1) s += __shfl_xor(s, off, 32);
    if (lane == 0) Erep[gw] = s;
}

// ---------------------------------------------------------------------------
// Stage 5 (one block per molecule): Loewdin generalized eigensolve.
//   eigh(S) -> phiS;  fockp = phiS^T F phiS (WMMA);  eigh(fockp) -> Eorb,C;
//   sort ascending; orb = phiS @ C_perm (WMMA);
//   rho_out = 2 * orb_occ @ orb_occ^T (WMMA, K=32);
//   Etot = sum(rho_out*H) + ener2 + Erep.
// ---------------------------------------------------------------------------
struct SolveShared {
    float M[BS * LDA];
    float V[BS * LDA];
    float P[BS * LDA];
    float evals[64];
    int   perm[64];
    int   idperm[64];
    int   top[32], bot[32], pp[32], qq[32];
    float cs[32], sn[32];
    float red[BLK];
};

__global__ __launch_bounds__(BLK)
void solve_kernel(const float* __restrict__ S, const float* __restrict__ Fws,
                  const float* __restrict__ Hws, const float* __restrict__ E2,
                  const float* __restrict__ Erep, float* __restrict__ out)
{
    __shared__ SolveShared L;
    const int mol = blockIdx.x;
    const int tid = threadIdx.x;
    const size_t mb = (size_t)mol * 4096;

    // load S into M
    for (int i = tid; i < 4096; i += BLK)
        L.M[(i >> 6) * LDA + (i & 63)] = S[mb + i];
    if (tid < 64) L.idperm[tid] = tid;
    __syncthreads();

    // eigh(S): S is diagonally dominant -> 6 sweeps
    jacobi64(L.M, L.V, L.top, L.bot, L.pp, L.qq, L.cs, L.sn, tid, 6);

    // phiS = Svecs * diag(Svals^-1/2)
    if (tid < 64) L.evals[tid] = rsqrtf(L.M[tid * LDA + tid]);
    __syncthreads();
    for (int i = tid; i < 4096; i += BLK) {
        int r = i >> 6, c = i & 63;
        L.P[r * LDA + c] = L.V[r * LDA + c] * L.evals[c];
    }
    __syncthreads();

    // load F into M
    for (int i = tid; i < 4096; i += BLK)
        L.M[(i >> 6) * LDA + (i & 63)] = Fws[mb + i];
    __syncthreads();

    // T = phiS^T @ F -> V
    gemm64<true, false>(L.P, L.M, L.V, 64, 1.0f, L.idperm, tid);
    __syncthreads();
    // fockp = T @ phiS -> M
    gemm64<false, false>(L.V, L.P, L.M, 64, 1.0f, L.idperm, tid);
    __syncthreads();

    // eigh(fockp): 10 sweeps
    jacobi64(L.M, L.V, L.top, L.bot, L.pp, L.qq, L.cs, L.sn, tid, 10);

    // extract eigenvalues, odd-even sort ascending with permutation
    if (tid < 64) { L.evals[tid] = L.M[tid * LDA + tid]; L.perm[tid] = tid; }
    __syncthreads();
    for (int it = 0; it < 64; ++it) {
        int off = it & 1;
        if (tid < 32) {
            int i = off + 2 * tid;
            if (i + 1 < 64 && L.evals[i] > L.evals[i + 1]) {
                float tv = L.evals[i]; L.evals[i] = L.evals[i + 1]; L.evals[i + 1] = tv;
                int tp = L.perm[i]; L.perm[i] = L.perm[i + 1]; L.perm[i + 1] = tp;
            }
        }
        __syncthreads();
    }

    // orb = phiS @ C[:, perm] -> M
    gemm64<false, false>(L.P, L.V, L.M, 64, 1.0f, L.perm, tid);
    __syncthreads();
    // rho_out = 2 * orb_occ @ orb_occ^T  (K = 32 occupied columns) -> V
    gemm64<false, true>(L.M, L.M, L.V, 32, 2.0f, L.idperm, tid);
    __syncthreads();

    // ener1 = sum(rho_out * H)
    float acc = 0.0f;
    for (int i = tid; i < 4096; i += BLK)
        acc += L.V[(i >> 6) * LDA + (i & 63)] * Hws[mb + i];
    L.red[tid] = acc;
    __syncthreads();
    for (int s2 = 128; s2 > 0; s2 >>= 1) {
        if (tid < s2) L.red[tid] += L.red[tid + s2];
        __syncthreads();
    }

    if (tid == 0) out[(size_t)mol * 65] = L.red[0] + E2[mol] + Erep[mol];
    if (tid < 64) out[(size_t)mol * 65 + 1 + tid] = L.evals[tid];
}

// ---------------------------------------------------------------------------
// Host-side launcher.  All kernels on `stream`; scratch carved from d_ws.
// ---------------------------------------------------------------------------
extern "C" void kernel_launch(void* const* d_in, const int* in_sizes, int n_in,
                              void* d_out, int out_size, void* d_ws, size_t ws_size,
                              hipStream_t stream)
{
    const float* A          = (const float*)d_in[0];
    const float* b          = (const float*)d_in[1];
    const float* coeffs     = (const float*)d_in[2];
    const float* rot_tensor = (const float*)d_in[3];
    const float* S          = (const float*)d_in[4];
    const float* G          = (const float*)d_in[5];
    const float* rho        = (const float*)d_in[6];
    const float* qneutral   = (const float*)d_in[7];
    const int*   g_rot_dir  = (const int*)d_in[9];
    const int*   g_rot      = (const int*)d_in[10];
    const int*   g_oper     = (const int*)d_in[11];
    const int*   g_rep      = (const int*)d_in[12];
    const int*   seg_ids    = (const int*)d_in[13];

    const int nsp     = in_sizes[1];
    const int nmol    = in_sizes[7] / 64;
    const int l0      = in_sizes[9];
    const int nrot    = in_sizes[10] / 3;
    const int nrep    = in_sizes[12];
    const int rot_len = 2 + l0 + nrot * 3;

    float* ws        = (float*)d_ws;
    float* net_vals  = ws;
    float* rot_out   = net_vals + nsp;
    float* Hws       = rot_out + (((size_t)rot_len + 3) & ~(size_t)3);
    float* Fws       = Hws + (size_t)nmol * 4096;
    float* E2        = Fws + (size_t)nmol * 4096;
    float* Erep      = E2 + nmol;

    spline_kernel<<<(nsp + 7) / 8, BLK, 0, stream>>>(A, b, coeffs, net_vals, nsp);

    int rtasks = l0 + nrot;
    rot_kernel<<<(rtasks + BLK - 1) / BLK, BLK, 0, stream>>>(
        net_vals, rot_tensor, g_rot_dir, g_rot, rot_out, l0, nrot);

    mull_kernel<<<nmol, BLK, 0, stream>>>(S, G, rho, qneutral, g_oper, rot_out,
                                          Fws, Hws, E2);

    erep_kernel<<<(nmol * 32 + BLK - 1) / BLK, BLK, 0, stream>>>(
        net_vals, g_rep, seg_ids, Erep, nrep, nmol);

    solve_kernel<<<nmol, BLK, 0, stream>>>(S, Fws, Hws, E2, Erep, (float*)d_out);
}